// HardAttentionLayer_45372034515616
// MI455X (gfx1250) — compile-verified
//
#include <hip/hip_runtime.h>
#include <math.h>

// ---------------------------------------------------------------------------
// HardAttentionLayer for MI455X (gfx1250).
// out[b,n,:] = x[b, argmax_s(logits[b,n,s]+g[b,n,s]), :]
// logits[b,n,s] = (x[b,s]+PE[s]) . v[b,n],  v[b,n] = Wq_n^T (Wk_n pesum[b])
// Bandwidth-bound: two streaming passes over x (2 x 419 MB ~ 36 us @ 23.3TB/s).
// Small GEMMs run on V_WMMA_F32_16X16X4_F32.
// ---------------------------------------------------------------------------

typedef __attribute__((ext_vector_type(2))) float v2f;
typedef __attribute__((ext_vector_type(8))) float v8f;

#define B_   1024
#define S_   100
#define H_   1024
#define NH_  4
#define A_   128
#define QK_  512    // NH_*A_

// workspace layout (float elements)
#define OFF_PE      0u          // PE matrix [S][H]            : 102400
#define OFF_PESUM   102400u     // sum_s PE[s][h]              : 1024
#define OFF_XSUM    103424u     // pesum[b][h]                 : 1048576
#define OFF_T       1152000u    // t[b][512]                   : 524288
#define OFF_V       1676288u    // v[b][n][h]                  : 4194304
#define OFF_SSTAR   5870592u    // argmax indices (int) [B*N]  : 4096

__device__ __forceinline__ v8f wmma_f32_16x16x4(v2f a, v2f b, v8f c) {
  // 8 args: (neg_a, A, neg_b, B, c_mod, C, reuse_a, reuse_b)
  return __builtin_amdgcn_wmma_f32_16x16x4_f32(false, a, false, b, (short)0, c,
                                               false, false);
}

// K0: sinusoidal PE [S][H] + its column sum [H]
__global__ void pe_kernel(float* __restrict__ pem, float* __restrict__ pesum) {
  int h = blockIdx.x * blockDim.x + threadIdx.x;
  if (h >= H_) return;
  int i2 = (h >> 1) << 1;  // 2*i
  float div = __expf((float)i2 * (-9.210340371976184f / (float)H_));
  float acc = 0.f;
  for (int s = 0; s < S_; ++s) {
    float ang = (float)s * div;
    float val = (h & 1) ? __cosf(ang) : __sinf(ang);
    pem[s * H_ + h] = val;
    acc += val;
  }
  pesum[h] = acc;
}

// K1: pesum[b][h] = sum_s x[b][s][h] + PEsum[h]   (streaming pass 1 over x)
__global__ void colsum_kernel(const float* __restrict__ x,
                              const float* __restrict__ pesum,
                              float* __restrict__ xsum) {
  int id = blockIdx.x * blockDim.x + threadIdx.x;  // b*H + h
  int h = id & (H_ - 1);
  const float* p = x + (size_t)(id >> 10) * (S_ * H_) + h;
  float acc = pesum[h];
#pragma unroll 4
  for (int s = 0; s < S_; ++s) acc += p[(size_t)s * H_];
  xsum[id] = acc;
}

// K2: t[B][512] = pesum[B][H] @ Wk^T   (WMMA, one wave per 16x16 tile)
__global__ void gemm_t_kernel(const float* __restrict__ xsum,
                              const float* __restrict__ Wk,
                              float* __restrict__ t) {
  int gtid = blockIdx.x * blockDim.x + threadIdx.x;
  int wave = gtid >> 5;
  int lane = threadIdx.x & 31;
  int nt = wave & 31;   // 32 N tiles (512/16)
  int mt = wave >> 5;   // 64 M tiles (1024/16)
  int l15 = lane & 15;
  int ks = (lane >> 4) << 1;
  const float* arow = xsum + (size_t)(mt * 16 + l15) * H_;  // A row m
  const float* brow = Wk + (size_t)(nt * 16 + l15) * H_;    // B[k][n]=Wk[n][k]
  v8f c = {};
#pragma unroll 8
  for (int k = 0; k < H_; k += 4) {
    v2f a = {arow[k + ks], arow[k + ks + 1]};
    v2f b = {brow[k + ks], brow[k + ks + 1]};
    c = wmma_f32_16x16x4(a, b, c);
  }
  int hi8 = (lane >> 4) << 3;
#pragma unroll
  for (int r = 0; r < 8; ++r)
    t[(size_t)(mt * 16 + r + hi8) * QK_ + nt * 16 + l15] = c[r];
}

// K2b: v[b][n][h] = sum_k t[b][n*128+k] * Wq[n*128+k][h]   (4 head GEMMs, WMMA)
__global__ void gemm_v_kernel(const float* __restrict__ t,
                              const float* __restrict__ Wq,
                              float* __restrict__ v) {
  int gtid = blockIdx.x * blockDim.x + threadIdx.x;
  int wave = gtid >> 5;
  int lane = threadIdx.x & 31;
  int n = wave >> 12;     // 4096 waves per head (64 x 64 tiles)
  int rem = wave & 4095;
  int mt = rem >> 6;      // 64 M tiles over b
  int ht = rem & 63;      // 64 N tiles over h
  int l15 = lane & 15;
  int ks = (lane >> 4) << 1;
  const float* arow = t + (size_t)(mt * 16 + l15) * QK_ + n * A_;
  const float* bcol = Wq + (size_t)(n * A_) * H_ + ht * 16 + l15;
  v8f c = {};
#pragma unroll 8
  for (int k = 0; k < A_; k += 4) {
    v2f a = {arow[k + ks], arow[k + ks + 1]};
    v2f b = {bcol[(size_t)(k + ks) * H_], bcol[(size_t)(k + ks + 1) * H_]};
    c = wmma_f32_16x16x4(a, b, c);
  }
  int hi8 = (lane >> 4) << 3;
#pragma unroll
  for (int r = 0; r < 8; ++r) {
    int row = mt * 16 + r + hi8;  // b
    v[((size_t)row * NH_ + n) * H_ + ht * 16 + l15] = c[r];
  }
}

// K3: per b: logits[s][n] = (x[b,s]+PE[s]).v[b,n] via WMMA (M=112 pad, N=16 pad,
// K=1024), then fused gumbel + argmax over s.  (streaming pass 2 over x)
__global__ void logits_argmax_kernel(const float* __restrict__ x,
                                     const float* __restrict__ pem,
                                     const float* __restrict__ vv,
                                     const float* __restrict__ gu,
                                     int* __restrict__ sstar) {
  __shared__ float slog[112][17];
  int b = blockIdx.x;
  int wave = threadIdx.x >> 5;
  int lane = threadIdx.x & 31;
  if (wave < 7) {  // 7 M tiles cover s=0..111
    int mt = wave;
    int l15 = lane & 15;
    int ks = (lane >> 4) << 1;
    int s = mt * 16 + l15;
    bool sv = (s < S_);
    bool nv = (l15 < NH_);
    const float* xrow = x + ((size_t)b * S_ + s) * H_;
    const float* perow = pem + (size_t)s * H_;
    const float* vrow = vv + ((size_t)b * NH_ + l15) * H_;
    v8f c = {};
    for (int kb = 0; kb < H_; kb += 128) {
      if (kb + 128 < H_) __builtin_prefetch(xrow + kb + 128, 0, 0);
#pragma unroll
      for (int kk = 0; kk < 128; kk += 4) {
        int k0 = kb + kk + ks;
        float a0 = sv ? (xrow[k0] + perow[k0]) : 0.f;
        float a1 = sv ? (xrow[k0 + 1] + perow[k0 + 1]) : 0.f;
        float b0 = nv ? vrow[k0] : 0.f;
        float b1 = nv ? vrow[k0 + 1] : 0.f;
        v2f a = {a0, a1};
        v2f bb = {b0, b1};
        c = wmma_f32_16x16x4(a, bb, c);
      }
    }
    int hi8 = (lane >> 4) << 3;
#pragma unroll
    for (int r = 0; r < 8; ++r) slog[mt * 16 + r + hi8][l15] = c[r];
  }
  __syncthreads();
  if (threadIdx.x < NH_) {
    int n = threadIdx.x;
    const float* gun = gu + ((size_t)b * NH_ + n) * S_;
    float best = -1e30f;
    int bi = 0;
    for (int s = 0; s < S_; ++s) {
      float u = gun[s];
      float g = -__logf(-__logf(u + 1e-10f) + 1e-10f);
      float val = slog[s][n] * (1.0f / 3200.0f) + g;  // 1/(sqrt(1024)*100)
      if (val > best) { best = val; bi = s; }
    }
    sstar[b * NH_ + n] = bi;
  }
}

// K4: out[b,n,:] = x[b, sstar[b,n], :]  (vectorized row gather)
__global__ void gather_kernel(const float* __restrict__ x,
                              const int* __restrict__ sstar,
                              float* __restrict__ out) {
  int id = blockIdx.x * blockDim.x + threadIdx.x;  // over B*N*H/4
  int bn = id >> 8;   // H/4 = 256 float4 per (b,n)
  int h4 = id & 255;
  int s = sstar[bn];
  int b = bn >> 2;
  const float4* src = (const float4*)(x + ((size_t)b * S_ + s) * H_);
  float4* dst = (float4*)(out + (size_t)bn * H_);
  dst[h4] = src[h4];
}

extern "C" void kernel_launch(void* const* d_in, const int* in_sizes, int n_in,
                              void* d_out, int out_size, void* d_ws,
                              size_t ws_size, hipStream_t stream) {
  (void)in_sizes; (void)n_in; (void)out_size; (void)ws_size;
  const float* x  = (const float*)d_in[0];
  const float* Wq = (const float*)d_in[1];
  const float* Wk = (const float*)d_in[2];
  const float* gu = (const float*)d_in[3];
  float* out = (float*)d_out;
  float* w = (float*)d_ws;

  float* pem    = w + OFF_PE;
  float* pesum  = w + OFF_PESUM;
  float* xsum   = w + OFF_XSUM;
  float* t      = w + OFF_T;
  float* v      = w + OFF_V;
  int*   sstar  = (int*)(w + OFF_SSTAR);

  pe_kernel<<<H_ / 256, 256, 0, stream>>>(pem, pesum);
  colsum_kernel<<<(B_ * H_) / 256, 256, 0, stream>>>(x, pesum, xsum);
  gemm_t_kernel<<<(64 * 32) / 8, 256, 0, stream>>>(xsum, Wk, t);       // 256 blk
  gemm_v_kernel<<<(NH_ * 64 * 64) / 8, 256, 0, stream>>>(t, Wq, v);    // 2048 blk
  logits_argmax_kernel<<<B_, 256, 0, stream>>>(x, pem, v, gu, sstar);
  gather_kernel<<<(B_ * NH_ * H_ / 4) / 256, 256, 0, stream>>>(x, sstar, out);
}